// LGMDNet_85332410237437
// MI455X (gfx1250) — compile-verified
//
#include <hip/hip_runtime.h>
#include <hip/hip_bf16.h>

// ---------------------------------------------------------------------------
// LGMD net, MI455X (gfx1250).
//  Problem is HBM-bandwidth bound (737 MB of x per call ~= 32us @ 23.3 TB/s).
//  Dependence restructuring:
//    k1: x -> exc           (parallel over T*B; the big streamer, pool+3x3+relu)
//    k2: exc_{t-1} -> lgmd  (parallel over T*B; 7x7 conv as Toeplitz GEMM on
//                            V_WMMA_F32_16X16X4_F32, + exc_t)     [WMMA path]
//    k3: per-pixel 64-step LIF recurrence in registers + dcmd reduction
// ---------------------------------------------------------------------------

#define T_STEPS 64
#define BATCH   16
#define HS      65
#define WSZ     86
#define IMG     (HS * WSZ)      // 5590
#define IN_H    260
#define IN_W    346

typedef __attribute__((ext_vector_type(2))) float v2f;
typedef __attribute__((ext_vector_type(8))) float v8f;

// ---------------------------------------------------------------------------
// Kernel 1: 4x4 avg pool + 3x3 conv (2 in-ch) + ReLU  ->  exc[t,b,65,86]
// One block = 13 output rows of one (t,b) image. 5*16*64 = 5120 blocks.
// ---------------------------------------------------------------------------
__global__ __launch_bounds__(256)
void k_pool_exc(const float* __restrict__ x, const float* __restrict__ exc_w,
                float* __restrict__ exc) {
    const int chunk = blockIdx.x;              // 0..4  (13 rows each, 5*13==65)
    const int b     = blockIdx.y;
    const int t     = blockIdx.z;
    const int tid   = threadIdx.x;
    const int r0    = chunk * 13;

    __shared__ float p_s[2][15][88];           // pooled rows r0-1..r0+13, halo'd cols

    for (int i = tid; i < 2 * 15 * 88; i += 256) ((float*)p_s)[i] = 0.0f;
    __syncthreads();

    const size_t img_off = (size_t)(t * BATCH + b) * 2 * IN_H * IN_W;

    // Pool: each pooled value = mean of a 4x4 block; rows via 8B-aligned v2f.
    for (int i = tid; i < 2 * 15 * WSZ; i += 256) {
        const int ch  = i / (15 * WSZ);
        const int rem = i % (15 * WSZ);
        const int lr  = rem / WSZ;
        const int pc  = rem % WSZ;
        const int pr  = r0 - 1 + lr;
        if (pr >= 0 && pr < HS) {
            const float* src = x + img_off + ((size_t)ch * IN_H + (size_t)pr * 4) * IN_W + pc * 4;
            __builtin_prefetch(src + 4 * IN_W, 0, 1);   // global_prefetch_b8
            float s = 0.0f;
            #pragma unroll
            for (int rr = 0; rr < 4; ++rr) {
                const v2f a = *(const v2f*)(src + rr * IN_W);
                const v2f c = *(const v2f*)(src + rr * IN_W + 2);
                s += (a.x + a.y) + (c.x + c.y);
            }
            p_s[ch][lr][pc + 1] = s * 0.0625f;
        }
    }

    float w[18];
    #pragma unroll
    for (int j = 0; j < 18; ++j) w[j] = exc_w[j];

    __syncthreads();

    float* dst = exc + (size_t)(t * BATCH + b) * IMG + (size_t)r0 * WSZ;
    for (int i = tid; i < 13 * WSZ; i += 256) {
        const int rl = i / WSZ;
        const int c  = i % WSZ;
        float acc = 0.0f;
        #pragma unroll
        for (int ch = 0; ch < 2; ++ch)
            #pragma unroll
            for (int kr = 0; kr < 3; ++kr)
                #pragma unroll
                for (int kc = 0; kc < 3; ++kc)
                    acc += p_s[ch][rl + kr][c + kc] * w[ch * 9 + kr * 3 + kc];
        dst[i] = fmaxf(acc, 0.0f);
    }
}

// ---------------------------------------------------------------------------
// Kernel 2: lgmd[t] = exc[t] + conv7x7(exc[t-1])   (exc[-1] == 0)
// One wave per 16x16 output tile. 7-tap row-conv per kernel-row kr is a
// Toeplitz GEMM: out(16x16) += A(16x22) * W_kr(22x16), W_kr[K][N]=w7[kr][K-N].
// K padded to 24 -> 6 x V_WMMA_F32_16X16X4_F32 per kr, 42 WMMAs per tile.
// B elements come from a zero-padded LDS table (branch-free ds_load_b32),
// indexed s_wpad[kr][ (K-N) + 16 ], nonzero only for K-N in [0,7).
// ---------------------------------------------------------------------------
__global__ __launch_bounds__(32)
void k_inh_wmma(const float* __restrict__ exc, const float* __restrict__ inh_w,
                float* __restrict__ lgmd) {
    const int tile = blockIdx.x;               // 0..29 (5 row-tiles x 6 col-tiles)
    const int b    = blockIdx.y;
    const int t    = blockIdx.z;
    const int lane = threadIdx.x;
    const int ty   = tile / 6, tx = tile % 6;
    const int r0   = ty * 16, c0 = tx * 16;

    __shared__ float s_in[22 * 24];            // input tile rows r0-3..r0+18, K-padded
    __shared__ float s_wpad[7 * 40];           // [kr][dx+16], dx in [-15,23], pad = 0

    // Zero-padded Toeplitz weight table (single wave: DS ops are in order,
    // barrier below covers visibility anyway).
    for (int i = lane; i < 7 * 40; i += 32) s_wpad[i] = 0.0f;
    for (int i = lane; i < 49; i += 32)
        s_wpad[(i / 7) * 40 + 16 + (i % 7)] = inh_w[i];

    if (t == 0) {                              // inh buffer starts at zero
        for (int i = lane; i < 22 * 24; i += 32) s_in[i] = 0.0f;
    } else {
        const float* prev = exc + (size_t)((t - 1) * BATCH + b) * IMG;
        for (int i = lane; i < 22 * 24; i += 32) {
            const int lr = i / 24, lc = i % 24;
            const int gr = r0 - 3 + lr, gc = c0 - 3 + lc;
            const bool ok = (gr >= 0) && (gr < HS) && (gc >= 0) && (gc < WSZ);
            const int cgr = min(max(gr, 0), HS - 1);     // clamp: load stays legal,
            const int cgc = min(max(gc, 0), WSZ - 1);    // select kills the halo
            const float v = prev[cgr * WSZ + cgc];
            s_in[i] = ok ? v : 0.0f;
        }
    }
    __syncthreads();

    const int n     = lane & 15;               // M for A, N for B, col for C/D
    const int half  = lane >> 4;
    const int bbase = 16 + 2 * half - n;       // s_wpad offset for ck=0 (.x elem)
    v8f acc = {};

    #pragma unroll
    for (int kr = 0; kr < 7; ++kr) {
        #pragma unroll
        for (int ck = 0; ck < 6; ++ck) {
            const int kbase = ck * 4 + 2 * half;   // global K of .x element
            v2f a, bv;
            a.x  = s_in[(kr + n) * 24 + kbase];            // ds_load_b64 (even idx)
            a.y  = s_in[(kr + n) * 24 + kbase + 1];
            bv.x = s_wpad[kr * 40 + bbase + ck * 4];       // branch-free Toeplitz
            bv.y = s_wpad[kr * 40 + bbase + ck * 4 + 1];
            acc = __builtin_amdgcn_wmma_f32_16x16x4_f32(
                false, a, false, bv, (short)0, acc, false, false);
        }
    }

    const size_t ib   = (size_t)(t * BATCH + b) * IMG;
    const float* exct = exc + ib;
    float*       dst  = lgmd + ib;
    #pragma unroll
    for (int i = 0; i < 8; ++i) {
        const int row = r0 + i + 8 * half;
        const int col = c0 + n;
        if (row < HS && col < WSZ)
            dst[row * WSZ + col] = acc[i] + exct[row * WSZ + col];
    }
}

// ---------------------------------------------------------------------------
// Kernel 3: per-pixel LIF recurrence over T in registers; spikes written in
// place over lgmd; dcmd[t,b] reduced via wave32 shuffles + one atomic.
// ---------------------------------------------------------------------------
__global__ void k_zero(float* __restrict__ out) {
    const int i = blockIdx.x * 256 + threadIdx.x;
    if (i < T_STEPS * BATCH) out[i] = 0.0f;
}

__global__ __launch_bounds__(256)
void k_recur(const float* __restrict__ dcmd_w, float* __restrict__ dcmd,
             float* __restrict__ spk /* lgmd in, spikes out (in place) */) {
    const int b      = blockIdx.y;
    const int tid    = threadIdx.x;
    const int idx    = blockIdx.x * 256 + tid;
    const bool valid = idx < IMG;
    const float aw   = valid ? fabsf(dcmd_w[idx]) : 0.0f;

    __shared__ float s_red[8];
    float v = 0.0f;

    for (int t = 0; t < T_STEPS; ++t) {
        float sp = 0.0f;
        const size_t off = (size_t)(t * BATCH + b) * IMG + idx;
        if (valid) {
            const float lin = spk[off];
            v += (lin - v) * 0.5f;             // TAU = 2
            sp = (v >= 1.0f) ? 1.0f : 0.0f;    // spike_fn(v - V_TH)
            spk[off] = sp;
            v *= (1.0f - sp);
        }
        float local = sp * aw;
        local += __shfl_down(local, 16);
        local += __shfl_down(local, 8);
        local += __shfl_down(local, 4);
        local += __shfl_down(local, 2);
        local += __shfl_down(local, 1);
        __syncthreads();                        // protect s_red across iterations
        if ((tid & 31) == 0) s_red[tid >> 5] = local;
        __syncthreads();
        if (tid == 0) {
            float s = 0.0f;
            #pragma unroll
            for (int wv = 0; wv < 8; ++wv) s += s_red[wv];
            atomicAdd(&dcmd[t * BATCH + b], s);
        }
    }
}

// ---------------------------------------------------------------------------
extern "C" void kernel_launch(void* const* d_in, const int* in_sizes, int n_in,
                              void* d_out, int out_size, void* d_ws, size_t ws_size,
                              hipStream_t stream) {
    const float* x      = (const float*)d_in[0];   // (64,16,2,260,346)
    const float* exc_w  = (const float*)d_in[1];   // (1,2,3,3)  -> 18
    const float* inh_w  = (const float*)d_in[2];   // (1,1,7,7)  -> 49
    const float* dcmd_w = (const float*)d_in[3];   // (1,1,65,86)-> 5590

    float* out  = (float*)d_out;
    float* dcmd = out;                             // (64,16)
    float* spk  = out + T_STEPS * BATCH;           // (64,16,1,65,86), lgmd staged here
    float* exc  = (float*)d_ws;                    // (64,16,65,86) = 22.9 MB

    k_zero<<<4, 256, 0, stream>>>(dcmd);

    dim3 g1(5, BATCH, T_STEPS);                    // 5120 blocks x 256
    k_pool_exc<<<g1, 256, 0, stream>>>(x, exc_w, exc);

    dim3 g2(30, BATCH, T_STEPS);                   // 30720 waves, 1 tile each
    k_inh_wmma<<<g2, 32, 0, stream>>>(exc, inh_w, spk);

    dim3 g3((IMG + 255) / 256, BATCH, 1);          // 22 x 16 blocks
    k_recur<<<g3, 256, 0, stream>>>(dcmd_w, dcmd, spk);
}